// SoftMoE6_3719441678648
// MI455X (gfx1250) — compile-verified
//
#include <hip/hip_runtime.h>
#include <math.h>

// SoftMoE block for MI455X (gfx1250). All GEMMs via V_WMMA_F32_16X16X4_F32,
// register-blocked MTxNT tiles/wave + block-shared A staged through LDS
// (320KB/WGP on CDNA5) to kill the 8x redundant A traffic per block.
// Shapes: B=256, N_IMG=128, D=768, E=3, S=44, H=1024.

#define B_    256
#define NIMG  128
#define TOK   129        // N_IMG + 1
#define D_    768
#define E_    3
#define S_    44
#define ES_   132        // E*S
#define ESP_  144        // ES padded to multiple of 16
#define H_    1024
#define HROWS 48         // S padded to multiple of 16
#define LN_EPS 1e-5f

typedef float v2f __attribute__((ext_vector_type(2)));
typedef float v8f __attribute__((ext_vector_type(8)));

// ---------------------------------------------------------------------------
// Direct-load variant (used only for K=132 stage).
// A frag (16x4 f32): lanes 0-15 VGPR0/1 = K 0/1, lanes 16-31 = K 2/3.
// B frag (4x16 f32): lanes 0-15 VGPR0/1 = rows K 0/1, lanes 16-31 = K 2/3.
// ---------------------------------------------------------------------------
template<int MT, int NT>
__device__ __forceinline__ void wmma_block(const float* __restrict__ A, int lda,
                                           const float* __restrict__ Bm, int ldb,
                                           int K, v8f acc[MT][NT]) {
  const int lane = threadIdx.x & 31;
  const int l16  = lane & 15;
  const int koff = (lane >> 4) << 1;               // 0 or 2
  const float* ap = A + (size_t)l16 * lda + koff;
  const float* bp = Bm + (size_t)koff * ldb + l16;
  for (int k = 0; k < K; k += 4) {
    v2f a[MT], b[NT];
#pragma unroll
    for (int i = 0; i < MT; ++i)
      a[i] = *(const v2f*)(ap + (size_t)(16 * i) * lda);
#pragma unroll
    for (int j = 0; j < NT; ++j) {
      b[j].x = bp[16 * j];
      b[j].y = bp[16 * j + ldb];
    }
#pragma unroll
    for (int i = 0; i < MT; ++i)
#pragma unroll
      for (int j = 0; j < NT; ++j)
        acc[i][j] = __builtin_amdgcn_wmma_f32_16x16x4_f32(
            false, a[i], false, b[j], (short)0, acc[i][j], false, false);
    ap += 4;
    bp += 4 * (size_t)ldb;
  }
}

// ---------------------------------------------------------------------------
// LDS-staged variant: the MT*16-row A macro-tile (shared by all 8 waves of the
// block) is staged chunk-by-chunk (KC wide) into LDS with coalesced float4
// loads; waves read A fragments from LDS (pitch KC+4 -> conflict-free b64
// reads), B comes straight from global (contiguous per half-wave).
// Double-buffered; one barrier per chunk; KC/4 * MT*NT WMMAs per chunk/wave.
// `active` must be wave-uniform; inactive waves still stage + barrier.
// ---------------------------------------------------------------------------
template<int MT, int NT, int KC>
__device__ __forceinline__ void wmma_block_lds(const float* __restrict__ A, int lda,
                                               const float* __restrict__ Bm, int ldb,
                                               int K, v8f acc[MT][NT],
                                               float* __restrict__ sm, bool active) {
  constexpr int ROWS  = MT * 16;
  constexpr int PITCH = KC + 4;          // 16B-aligned rows, conflict-free frags
  constexpr int BUFSZ = ROWS * PITCH;
  constexpr int F4S   = ROWS * (KC / 4); // float4s per chunk
  const int t    = (int)threadIdx.x;     // 0..255
  const int lane = t & 31;
  const int l16  = lane & 15;
  const int koff = (lane >> 4) << 1;

  auto stage = [&](int k0, int buf) {
    float* dst = sm + buf * BUFSZ;
    for (int i = t; i < F4S; i += 256) {
      const int row = i / (KC / 4);
      const int c4  = i % (KC / 4);
      *(float4*)(dst + row * PITCH + c4 * 4) =
          *(const float4*)(A + (size_t)row * lda + k0 + c4 * 4);
    }
  };

  stage(0, 0);
  __syncthreads();
  const float* bp = Bm + (size_t)koff * ldb + l16;
  const int nchunks = K / KC;
  for (int c = 0; c < nchunks; ++c) {
    if (c + 1 < nchunks) stage((c + 1) * KC, (c + 1) & 1);
    const float* abase = sm + (c & 1) * BUFSZ + koff;
    if (active) {
#pragma unroll
      for (int kc = 0; kc < KC; kc += 4) {
        v2f a[MT], b[NT];
#pragma unroll
        for (int i = 0; i < MT; ++i)
          a[i] = *(const v2f*)(abase + (16 * i + l16) * PITCH + kc);
#pragma unroll
        for (int j = 0; j < NT; ++j) {
          b[j].x = bp[16 * j];
          b[j].y = bp[16 * j + ldb];
        }
#pragma unroll
        for (int i = 0; i < MT; ++i)
#pragma unroll
          for (int j = 0; j < NT; ++j)
            acc[i][j] = __builtin_amdgcn_wmma_f32_16x16x4_f32(
                false, a[i], false, b[j], (short)0, acc[i][j], false, false);
        bp += 4 * (size_t)ldb;
      }
    }
    __syncthreads();
  }
}

template<int MT, int NT>
__device__ __forceinline__ void zero_acc(v8f acc[MT][NT]) {
#pragma unroll
  for (int i = 0; i < MT; ++i)
#pragma unroll
    for (int j = 0; j < NT; ++j)
      acc[i][j] = {};
}

// Epilogue functor receives (row, col, value) in matrix coordinates.
template<int MT, int NT, typename Epi>
__device__ __forceinline__ void store_tiles(v8f acc[MT][NT], int tileM, int tileN,
                                            Epi epi) {
  const int lane = threadIdx.x & 31, l16 = lane & 15, rb = (lane >> 4) * 8;
#pragma unroll
  for (int i = 0; i < MT; ++i)
#pragma unroll
    for (int j = 0; j < NT; ++j) {
      const int col = tileN + 16 * j + l16;
#pragma unroll
      for (int r = 0; r < 8; ++r)
        epi(tileM + 16 * i + rb + r, col, acc[i][j][r]);
    }
}

__device__ __forceinline__ float gelu_f(float x) {
  return 0.5f * x * (1.0f + erff(x * 0.7071067811865476f));
}

// ---------------------------------------------------------------------------
// Small prep kernels
// ---------------------------------------------------------------------------
__global__ void k_attnfeat(const float* __restrict__ aw, float* __restrict__ af) {
  __shared__ float sm[128];
  const int b = blockIdx.x, t = threadIdx.x;
  sm[t] = aw[b * NIMG + t];
  __syncthreads();
  for (int s = 64; s > 0; s >>= 1) { if (t < s) sm[t] += sm[t + s]; __syncthreads(); }
  if (t == 0) af[b] = sm[0] * (1.0f / NIMG);
}

// LayerNorm of img tokens (scaled by `scale`) + per-token expert id.
__global__ void k_prep(const float* __restrict__ x, const float* __restrict__ aw,
                       const float* __restrict__ gamma, const float* __restrict__ beta,
                       const float* __restrict__ scale,
                       float* __restrict__ img_ns, int* __restrict__ eid) {
  __shared__ float s1[256], s2[256];
  const int row = blockIdx.x;              // b*NIMG + n
  const int b = row / NIMG, n = row % NIMG;
  const int t = threadIdx.x;
  const float* xr = x + ((size_t)b * TOK + 1 + n) * D_;
  float s = 0.f, ss = 0.f;
  for (int i = t; i < D_; i += 256) { float v = xr[i]; s += v; ss += v * v; }
  s1[t] = s; s2[t] = ss; __syncthreads();
  for (int st = 128; st > 0; st >>= 1) {
    if (t < st) { s1[t] += s1[t + st]; s2[t] += s2[t + st]; }
    __syncthreads();
  }
  const float mean = s1[0] * (1.0f / D_);
  const float var  = s2[0] * (1.0f / D_) - mean * mean;
  const float rstd = rsqrtf(var + LN_EPS);
  const float sc   = scale[0];
  float* o = img_ns + (size_t)row * D_;
  for (int i = t; i < D_; i += 256)
    o[i] = ((xr[i] - mean) * rstd * gamma[i] + beta[i]) * sc;
  if (t == 0) {
    const float a = aw[b * NIMG + n];
    eid[row] = (a > 0.7f) ? 0 : ((a >= 0.3f) ? 1 : 2);  // vis / part / occ
  }
}

// mu (E,D,88) -> mu_t (D, ESP) with cols >= ES zeroed.
__global__ void k_mut(const float* __restrict__ mu, float* __restrict__ mu_t) {
  const int d = blockIdx.x, es = threadIdx.x;
  float v = 0.f;
  if (es < ES_) {
    const int e = es / S_, s = es % S_;
    v = mu[((size_t)e * D_ + d) * 88 + s];
  }
  mu_t[(size_t)d * ESP_ + es] = v;
}

// rowdot[b,n] = (scale*img_n[b,n,:]) . slot_bias[b,:]
__global__ void k_rowdot(const float* __restrict__ img_ns, const float* __restrict__ sb,
                         float* __restrict__ rd) {
  __shared__ float sm[256];
  const int row = blockIdx.x, b = row / NIMG, t = threadIdx.x;
  const float* a = img_ns + (size_t)row * D_;
  const float* w = sb + (size_t)b * D_;
  float s = 0.f;
  for (int i = t; i < D_; i += 256) s += a[i] * w[i];
  sm[t] = s; __syncthreads();
  for (int st = 128; st > 0; st >>= 1) { if (t < st) sm[t] += sm[t + st]; __syncthreads(); }
  if (t == 0) rd[row] = sm[0];
}

// ---------------------------------------------------------------------------
// WMMA GEMM stages. Block = 256 threads = 8 waves; wave w owns macro-tile
// columns [ (bx*8+w)*NT*16 , +NT*16 ), rows [ by*MT*16 , +MT*16 ).
// A macro-tile is identical for all 8 waves -> staged via LDS.
// ---------------------------------------------------------------------------

// slot_bias = cls @ vsg_w[:768] + attn_feat*vsg_w[768] + vsg_b (M=256,N=768,K=768)
__global__ void __launch_bounds__(256) k_slotbias(
    const float* __restrict__ x, const float* __restrict__ vw,
    const float* __restrict__ vb, const float* __restrict__ af,
    float* __restrict__ sb) {
  constexpr int MT = 4, NT = 2, KC = 32;
  __shared__ float sm[2 * MT * 16 * (KC + 4)];
  const int wave = threadIdx.x >> 5;
  const int tileN = (blockIdx.x * 8 + wave) * (NT * 16);
  const int tileM = blockIdx.y * (MT * 16);
  v8f acc[MT][NT];
  zero_acc<MT, NT>(acc);
  wmma_block_lds<MT, NT, KC>(x + (size_t)tileM * (TOK * D_), TOK * D_,
                             vw + tileN, D_, D_, acc, sm, true);
  store_tiles<MT, NT>(acc, tileM, tileN, [=](int row, int col, float v) {
    sb[(size_t)row * D_ + col] = v + af[row] * vw[(size_t)D_ * D_ + col] + vb[col];
  });
}

// logits[b,n,es] = img_ns . mu_t[:,es] + rowdot[b,n], expert-masked (M=128,N=132,K=768)
__global__ void __launch_bounds__(256) k_logits(
    const float* __restrict__ img_ns, const float* __restrict__ mu_t,
    const float* __restrict__ rd, const int* __restrict__ eid,
    float* __restrict__ logits) {
  constexpr int MT = 4, NT = 1, KC = 32;
  __shared__ float sm[2 * MT * 16 * (KC + 4)];
  const int wave = threadIdx.x >> 5;
  const int tileN = (blockIdx.x * 8 + wave) * (NT * 16);
  const bool active = (tileN < ESP_);              // wave-uniform
  const int tileM = blockIdx.y * (MT * 16);
  const int b = blockIdx.z;
  v8f acc[MT][NT];
  zero_acc<MT, NT>(acc);
  wmma_block_lds<MT, NT, KC>(img_ns + ((size_t)b * NIMG + tileM) * D_, D_,
                             mu_t + (active ? tileN : 0), ESP_, D_, acc, sm, active);
  if (active)
    store_tiles<MT, NT>(acc, tileM, tileN, [=](int row, int col, float v) {
      if (col < ES_) {
        const int g = b * NIMG + row;
        float o = v + rd[g];
        o = (eid[g] == (col / S_)) ? o : 0.0f;
        logits[((size_t)b * NIMG + row) * ES_ + col] = o;
      }
    });
}

// dispatch softmax over n; store transposed (B, ESP, NIMG) for slot_in GEMM.
__global__ void k_dispatch(const float* __restrict__ logits, float* __restrict__ dispT) {
  __shared__ float sm[128];
  const int idx = blockIdx.x, b = idx / ES_, es = idx % ES_, t = threadIdx.x;
  const float v = logits[((size_t)b * NIMG + t) * ES_ + es];
  sm[t] = v; __syncthreads();
  for (int st = 64; st > 0; st >>= 1) { if (t < st) sm[t] = fmaxf(sm[t], sm[t + st]); __syncthreads(); }
  const float mx = sm[0]; __syncthreads();
  const float e = __expf(v - mx);
  sm[t] = e; __syncthreads();
  for (int st = 64; st > 0; st >>= 1) { if (t < st) sm[t] += sm[t + st]; __syncthreads(); }
  dispT[((size_t)b * ESP_ + es) * NIMG + t] = e / sm[0];
}

// combine softmax over 132 slots per token; store (B, NIMG, ES).
__global__ void k_combine(const float* __restrict__ logits, float* __restrict__ comb) {
  __shared__ float sm[256];
  const int idx = blockIdx.x, t = threadIdx.x;
  const float* lr = logits + (size_t)idx * ES_;
  const float v = (t < ES_) ? lr[t] : -1e30f;
  sm[t] = v; __syncthreads();
  for (int st = 128; st > 0; st >>= 1) { if (t < st) sm[t] = fmaxf(sm[t], sm[t + st]); __syncthreads(); }
  const float mx = sm[0]; __syncthreads();
  const float e = (t < ES_) ? __expf(v - mx) : 0.f;
  sm[t] = e; __syncthreads();
  for (int st = 128; st > 0; st >>= 1) { if (t < st) sm[t] += sm[t + st]; __syncthreads(); }
  if (t < ES_) comb[(size_t)idx * ES_ + t] = e / sm[0];
}

// slot_in[b,es,d] = dispatchT[b,es,:] @ img[b,:,d]   (M=132->144,N=768,K=128)
__global__ void __launch_bounds__(256) k_slotin(
    const float* __restrict__ dispT, const float* __restrict__ x,
    float* __restrict__ slot_in) {
  constexpr int MT = 3, NT = 2, KC = 32;
  __shared__ float sm[2 * MT * 16 * (KC + 4)];
  const int wave = threadIdx.x >> 5;
  const int tileN = (blockIdx.x * 8 + wave) * (NT * 16);
  const int tileM = blockIdx.y * (MT * 16);
  const int b = blockIdx.z;
  v8f acc[MT][NT];
  zero_acc<MT, NT>(acc);
  wmma_block_lds<MT, NT, KC>(dispT + ((size_t)b * ESP_ + tileM) * NIMG, NIMG,
                             x + ((size_t)b * TOK + 1) * D_ + tileN, D_, NIMG,
                             acc, sm, true);
  store_tiles<MT, NT>(acc, tileM, tileN, [=](int row, int col, float v) {
    if (row < ES_)
      slot_in[((size_t)b * ESP_ + row) * D_ + col] = v;
  });
}

// h = gelu(slot_in[b,e] @ w1[e] + b1[e])   (M=44->48,N=1024,K=768), z = b*E+e
__global__ void __launch_bounds__(256) k_w1(
    const float* __restrict__ slot_in, const float* __restrict__ w1,
    const float* __restrict__ b1, float* __restrict__ h) {
  constexpr int MT = 3, NT = 2, KC = 32;
  __shared__ float sm[2 * MT * 16 * (KC + 4)];
  const int wave = threadIdx.x >> 5;
  const int tileN = (blockIdx.x * 8 + wave) * (NT * 16);
  const int tileM = blockIdx.y * (MT * 16);
  const int z = blockIdx.z, b = z / E_, e = z % E_;
  v8f acc[MT][NT];
  zero_acc<MT, NT>(acc);
  wmma_block_lds<MT, NT, KC>(slot_in + ((size_t)b * ESP_ + e * S_ + tileM) * D_, D_,
                             w1 + (size_t)e * D_ * H_ + tileN, H_, D_, acc, sm, true);
  store_tiles<MT, NT>(acc, tileM, tileN, [=](int row, int col, float v) {
    if (row < S_)
      h[((size_t)z * HROWS + row) * H_ + col] = gelu_f(v + b1[e * H_ + col]);
  });
}

// slot_out[b,e] = h[b,e] @ w2[e] + b2[e]   (M=44->48,N=768,K=1024)
__global__ void __launch_bounds__(256) k_w2(
    const float* __restrict__ h, const float* __restrict__ w2,
    const float* __restrict__ b2, float* __restrict__ slot_out) {
  constexpr int MT = 3, NT = 2, KC = 32;
  __shared__ float sm[2 * MT * 16 * (KC + 4)];
  const int wave = threadIdx.x >> 5;
  const int tileN = (blockIdx.x * 8 + wave) * (NT * 16);
  const int tileM = blockIdx.y * (MT * 16);
  const int z = blockIdx.z, b = z / E_, e = z % E_;
  v8f acc[MT][NT];
  zero_acc<MT, NT>(acc);
  wmma_block_lds<MT, NT, KC>(h + ((size_t)z * HROWS + tileM) * H_, H_,
                             w2 + (size_t)e * H_ * D_ + tileN, D_, H_, acc, sm, true);
  store_tiles<MT, NT>(acc, tileM, tileN, [=](int row, int col, float v) {
    if (row < S_)
      slot_out[((size_t)b * ESP_ + e * S_ + row) * D_ + col] = v + b2[e * D_ + col];
  });
}

// img_out[b,n,:] = combine[b,n,:] @ slot_out[b]   (M=128,N=768,K=132)
// K=132 not a multiple of KC -> direct-load path (only ~5% of total FLOPs).
__global__ void __launch_bounds__(256) k_imgout(
    const float* __restrict__ comb, const float* __restrict__ slot_out,
    float* __restrict__ out) {
  constexpr int MT = 4, NT = 2;
  const int wave = threadIdx.x >> 5;
  const int tileN = (blockIdx.x * 8 + wave) * (NT * 16);
  const int tileM = blockIdx.y * (MT * 16);
  const int b = blockIdx.z;
  v8f acc[MT][NT];
  zero_acc<MT, NT>(acc);
  wmma_block<MT, NT>(comb + ((size_t)b * NIMG + tileM) * ES_, ES_,
                     slot_out + (size_t)b * ESP_ * D_ + tileN, D_, ES_, acc);
  store_tiles<MT, NT>(acc, tileM, tileN, [=](int row, int col, float v) {
    out[((size_t)b * TOK + 1 + row) * D_ + col] = v;
  });
}

// cls_h = gelu(cls @ cw1 + cb1)   (M=256,N=3072,K=768)
__global__ void __launch_bounds__(256) k_cls1(
    const float* __restrict__ x, const float* __restrict__ cw1,
    const float* __restrict__ cb1, float* __restrict__ cls_h) {
  constexpr int MT = 4, NT = 2, KC = 32;
  __shared__ float sm[2 * MT * 16 * (KC + 4)];
  const int wave = threadIdx.x >> 5;
  const int tileN = (blockIdx.x * 8 + wave) * (NT * 16);
  const int tileM = blockIdx.y * (MT * 16);
  v8f acc[MT][NT];
  zero_acc<MT, NT>(acc);
  wmma_block_lds<MT, NT, KC>(x + (size_t)tileM * (TOK * D_), TOK * D_,
                             cw1 + tileN, 4 * D_, D_, acc, sm, true);
  store_tiles<MT, NT>(acc, tileM, tileN, [=](int row, int col, float v) {
    cls_h[(size_t)row * (4 * D_) + col] = gelu_f(v + cb1[col]);
  });
}

// cls_out = cls_h @ cw2 + cb2 -> out[:,0,:]   (M=256,N=768,K=3072)
__global__ void __launch_bounds__(256) k_cls2(
    const float* __restrict__ cls_h, const float* __restrict__ cw2,
    const float* __restrict__ cb2, float* __restrict__ out) {
  constexpr int MT = 4, NT = 2, KC = 32;
  __shared__ float sm[2 * MT * 16 * (KC + 4)];
  const int wave = threadIdx.x >> 5;
  const int tileN = (blockIdx.x * 8 + wave) * (NT * 16);
  const int tileM = blockIdx.y * (MT * 16);
  v8f acc[MT][NT];
  zero_acc<MT, NT>(acc);
  wmma_block_lds<MT, NT, KC>(cls_h + (size_t)tileM * (4 * D_), 4 * D_,
                             cw2 + tileN, D_, 4 * D_, acc, sm, true);
  store_tiles<MT, NT>(acc, tileM, tileN, [=](int row, int col, float v) {
    out[(size_t)row * TOK * D_ + col] = v + cb2[col];
  });
}

// ---------------------------------------------------------------------------
extern "C" void kernel_launch(void* const* d_in, const int* in_sizes, int n_in,
                              void* d_out, int out_size, void* d_ws, size_t ws_size,
                              hipStream_t stream) {
  const float* x   = (const float*)d_in[0];
  const float* aw  = (const float*)d_in[1];
  const float* gam = (const float*)d_in[2];
  const float* bet = (const float*)d_in[3];
  const float* vw  = (const float*)d_in[4];
  const float* vb  = (const float*)d_in[5];
  const float* mu  = (const float*)d_in[6];
  const float* sc  = (const float*)d_in[7];
  const float* w1  = (const float*)d_in[8];
  const float* b1  = (const float*)d_in[9];
  const float* w2  = (const float*)d_in[10];
  const float* b2  = (const float*)d_in[11];
  const float* cw1 = (const float*)d_in[12];
  const float* cb1 = (const float*)d_in[13];
  const float* cw2 = (const float*)d_in[14];
  const float* cb2 = (const float*)d_in[15];
  float* out = (float*)d_out;
  float* ws  = (float*)d_ws;

  // Workspace layout (floats). Total ~134.1M floats (~537 MB).
  constexpr size_t O_IMGNS = 0;                                       // B*NIMG*D
  constexpr size_t O_MUT   = O_IMGNS + (size_t)B_ * NIMG * D_;        // D*ESP
  constexpr size_t O_SB    = O_MUT   + (size_t)D_ * ESP_;             // B*D
  constexpr size_t O_AF    = O_SB    + (size_t)B_ * D_;               // B
  constexpr size_t O_RD    = O_AF    + (size_t)B_;                    // B*NIMG
  constexpr size_t O_LOG   = O_RD    + (size_t)B_ * NIMG;             // B*NIMG*ES
  constexpr size_t O_DT    = O_LOG   + (size_t)B_ * NIMG * ES_;       // B*ESP*NIMG
  constexpr size_t O_SIN   = O_DT    + (size_t)B_ * ESP_ * NIMG;      // B*ESP*D
  constexpr size_t O_H     = O_SIN   + (size_t)B_ * ESP_ * D_;        // B*E*HROWS*H
  constexpr size_t O_SOUT  = O_H     + (size_t)B_ * E_ * HROWS * H_;  // B*ESP*D
  constexpr size_t O_CMB   = O_SOUT  + (size_t)B_ * ESP_ * D_;        // B*NIMG*ES
  constexpr size_t O_CLSH  = O_CMB   + (size_t)B_ * NIMG * ES_;       // B*4D
  constexpr size_t O_EID   = O_CLSH  + (size_t)B_ * 4 * D_;           // B*NIMG ints

  float* img_ns = ws + O_IMGNS;
  float* mu_t   = ws + O_MUT;
  float* sb     = ws + O_SB;
  float* af     = ws + O_AF;
  float* rd     = ws + O_RD;
  float* logits = ws + O_LOG;
  float* dispT  = ws + O_DT;
  float* s_in   = ws + O_SIN;
  float* hbuf   = ws + O_H;
  float* s_out  = ws + O_SOUT;
  float* comb   = ws + O_CMB;
  float* cls_h  = ws + O_CLSH;
  int*   eid    = (int*)(ws + O_EID);

  k_attnfeat<<<B_, 128, 0, stream>>>(aw, af);
  k_prep<<<B_ * NIMG, 256, 0, stream>>>(x, aw, gam, bet, sc, img_ns, eid);
  k_mut<<<D_, ESP_, 0, stream>>>(mu, mu_t);
  // MT=4,NT=2: block covers 64 x 256
  k_slotbias<<<dim3(3, 4, 1), 256, 0, stream>>>(x, vw, vb, af, sb);
  k_rowdot<<<B_ * NIMG, 256, 0, stream>>>(img_ns, sb, rd);
  // MT=4,NT=1: block covers 64 x 128; N=144 padded -> grid.x=2 with inactive waves
  k_logits<<<dim3(2, 2, B_), 256, 0, stream>>>(img_ns, mu_t, rd, eid, logits);
  k_dispatch<<<B_ * ES_, 128, 0, stream>>>(logits, dispT);
  k_combine<<<B_ * NIMG, 256, 0, stream>>>(logits, comb);
  // MT=3,NT=2: block covers 48 x 256
  k_slotin<<<dim3(3, 3, B_), 256, 0, stream>>>(dispT, x, s_in);
  k_w1<<<dim3(4, 1, B_ * E_), 256, 0, stream>>>(s_in, w1, b1, hbuf);
  k_w2<<<dim3(3, 1, B_ * E_), 256, 0, stream>>>(hbuf, w2, b2, s_out);
  // MT=4,NT=2 (direct loads, K=132)
  k_imgout<<<dim3(3, 2, B_), 256, 0, stream>>>(comb, s_out, out);
  k_cls1<<<dim3(12, 4, 1), 256, 0, stream>>>(x, cw1, cb1, cls_h);
  k_cls2<<<dim3(3, 4, 1), 256, 0, stream>>>(cls_h, cw2, cb2, out);
}